// Enframe_16071767622237
// MI455X (gfx1250) — compile-verified
//
#include <hip/hip_runtime.h>

// Enframe: out[b, w, f] = in[b, f*HOP + w]
// in:  (16, 480000) f32     out: (16, 2048, 934) f32
#define FRAME_LEN 2048
#define HOP       512
#define BATCH     16
#define SAMPLES   480000
#define FRAMES    934                         // (480000-2048)/512 + 1
#define F_TILE    16                          // frames per block
#define SPAN      ((F_TILE - 1) * HOP + FRAME_LEN)   // 9728 floats staged per block
#define PAD4(i)   ((i) + (((i) >> 9) << 2))   // +4 floats every 512 -> conflict-free, keeps 16B align
#define LDS_FLOATS (SPAN + (SPAN / 512) * 4)  // 9728 + 76 = 9804 floats (~38.3 KB)
#define NTHREADS  256

typedef int   v4i __attribute__((ext_vector_type(4)));
typedef float v2f __attribute__((ext_vector_type(2)));
#define AS_GLOBAL __attribute__((address_space(1)))
#define AS_LDS    __attribute__((address_space(3)))

#if defined(__has_builtin)
#  if __has_builtin(__builtin_amdgcn_global_load_async_to_lds_b128)
#    define USE_ASYNC_LDS 1
#  endif
#endif

__global__ __launch_bounds__(NTHREADS)
void enframe_kernel(const float* __restrict__ in, float* __restrict__ out) {
    __shared__ float lds[LDS_FLOATS];

    const int tid = threadIdx.x;
    const int b   = blockIdx.y;
    const int f0  = blockIdx.x * F_TILE;               // first frame of this tile
    const int span_start = f0 * HOP;                   // input float index of span start
    const int span_valid = min(SPAN, SAMPLES - span_start);  // multiple of 4
    const float* __restrict__ src = in + (size_t)b * SAMPLES + span_start;

    // ---- Stage contiguous input span into LDS (16B async chunks, padded layout) ----
    for (int c = tid; c < SPAN / 4; c += NTHREADS) {
        const int i = c << 2;                          // float offset within span
        if (i < span_valid) {
            const int li = PAD4(i);                    // 16B-aligned (i%4==0, pad=16B)
#if USE_ASYNC_LDS
            __builtin_amdgcn_global_load_async_to_lds_b128(
                (AS_GLOBAL v4i*)(src + i),
                (AS_LDS    v4i*)(&lds[li]),
                0, 0);
#else
            const float4 v = *(const float4*)(src + i);
            *(float4*)(&lds[li]) = v;
#endif
        }
    }
#if USE_ASYNC_LDS
#  if defined(__has_builtin) && __has_builtin(__builtin_amdgcn_s_wait_asynccnt)
    __builtin_amdgcn_s_wait_asynccnt(0);
#  else
    asm volatile("s_wait_asynccnt 0" ::: "memory");
#  endif
#endif
    __syncthreads();

    // ---- Emit outputs: lanes cover consecutive f (coalesced), 8B NT stores ----
    // thread -> (fl = 2*(tid&7), w = tid>>3 + 32*k); wave covers 4 w-rows x 16 f (64B runs)
    const int fl = (tid & 7) << 1;                     // 0,2,...,14
    const int fv = min(F_TILE, FRAMES - f0);           // frames valid in tile (always even)
    float* __restrict__ dstb =
        out + (size_t)b * FRAME_LEN * FRAMES + (size_t)f0 + fl;

    if (fl < fv) {
#pragma unroll 4
        for (int w = tid >> 3; w < FRAME_LEN; w += NTHREADS / 8) {
            v2f v;
            v.x = lds[PAD4(fl * HOP + w)];
            v.y = lds[PAD4((fl + 1) * HOP + w)];
            __builtin_nontemporal_store(
                v, (v2f*)(dstb + (size_t)w * FRAMES));  // 8B aligned: f0+fl, FRAMES even
        }
    }
}

extern "C" void kernel_launch(void* const* d_in, const int* in_sizes, int n_in,
                              void* d_out, int out_size, void* d_ws, size_t ws_size,
                              hipStream_t stream) {
    (void)in_sizes; (void)n_in; (void)d_ws; (void)ws_size; (void)out_size;
    const float* in  = (const float*)d_in[0];
    float*       out = (float*)d_out;

    dim3 grid((FRAMES + F_TILE - 1) / F_TILE, BATCH);  // (59, 16)
    dim3 block(NTHREADS);
    enframe_kernel<<<grid, block, 0, stream>>>(in, out);
}